// TopKSAE_10591389352348
// MI455X (gfx1250) — compile-verified
//
#include <hip/hip_runtime.h>
#include <hip/hip_bf16.h>
#include <math.h>

typedef __attribute__((ext_vector_type(2))) float v2f;
typedef __attribute__((ext_vector_type(8))) float v8f;
typedef int v4i __attribute__((vector_size(16)));   // GCC-style, matches builtin

#define NROWS 8192
#define DIM   768
#define HID   16384
#define TOPK  32

#define AS_GLOBAL __attribute__((address_space(1)))
#define AS_LDS    __attribute__((address_space(3)))

#if __has_builtin(__builtin_amdgcn_global_load_async_to_lds_b128)
#define USE_ASYNC_LDS 1
#else
#define USE_ASYNC_LDS 0
#endif

#if USE_ASYNC_LDS
__device__ __forceinline__ void async_copy16(const void* g, void* l) {
  __builtin_amdgcn_global_load_async_to_lds_b128(
      (AS_GLOBAL v4i*)g, (AS_LDS v4i*)l, 0, 0);
}
#endif

__device__ __forceinline__ void async_wait0() {
#if __has_builtin(__builtin_amdgcn_s_wait_asynccnt)
  __builtin_amdgcn_s_wait_asynccnt(0);
#else
  asm volatile("s_wait_asynccnt 0x0" ::: "memory");
#endif
}

// ---------------------------------------------------------------------------
// Kernel 1: encode GEMM  pre = x @ W_enc + b_enc   (f32 WMMA 16x16x4)
// Block: 256 threads = 8 waves (2x4 wave grid), block tile 32(M) x 128(N),
// each wave owns a 16x32 output (2 accumulators -> each A fragment feeds 2
// WMMAs). K staged through LDS in chunks of 32 via async-to-LDS copies.
// pre is written with non-temporal stores (512MB stream, read-once later;
// keeps the 48MB W_enc resident in the 192MB L2).
// ---------------------------------------------------------------------------
__global__ __launch_bounds__(256)
void sae_encode_wmma(const float* __restrict__ x,
                     const float* __restrict__ W_enc,
                     const float* __restrict__ b_enc,
                     float* __restrict__ pre) {
  __shared__ float xs[32 * 36];    // 32 M x 32 K   (stride 36: 144B, 16B-aligned)
  __shared__ float wsd[32 * 132];  // 32 K x 128 N  (stride 132: 528B, 16B-aligned)

  const int t    = threadIdx.x;
  const int lane = t & 31;
  const int wave = t >> 5;
  const int wm   = wave >> 2;     // 0..1 : wave row
  const int wn   = wave & 3;      // 0..3 : wave col

  const int m0 = blockIdx.y * 32;
  const int n0 = blockIdx.x * 128;

  // A fragment (ISA 7.12.2): lane L -> M = L%16, K pair (L/16)*2
  const int am    = 16 * wm + (lane & 15);
  const int kbase = (lane >> 4) * 2;
  // B fragments: lane L -> N = L%16, K rows split at half-wave
  const int bn0   = 16 * wn + (lane & 15);
  const int bn1   = bn0 + 64;

  v8f acc0 = {0.f, 0.f, 0.f, 0.f, 0.f, 0.f, 0.f, 0.f};
  v8f acc1 = {0.f, 0.f, 0.f, 0.f, 0.f, 0.f, 0.f, 0.f};

  for (int kc = 0; kc < DIM; kc += 32) {
#if USE_ASYNC_LDS
    {
      // x tile: 32 rows x 32 K = 256 x 16B chunks, one async b128 per thread
      int r  = t >> 3;
      int c4 = (t & 7) * 4;
      async_copy16(x + (size_t)(m0 + r) * DIM + kc + c4, &xs[r * 36 + c4]);
      // W_enc tile: 32 K x 128 N = 1024 x 16B chunks, 4 per thread
      #pragma unroll
      for (int j = 0; j < 4; ++j) {
        int i   = t + 256 * j;
        int rr  = i >> 5;
        int cc4 = (i & 31) * 4;
        async_copy16(W_enc + (size_t)(kc + rr) * HID + n0 + cc4,
                     &wsd[rr * 132 + cc4]);
      }
      async_wait0();
    }
#else
    #pragma unroll
    for (int j = 0; j < 4; ++j) {
      int i = t + 256 * j;
      int r = i >> 5, c = i & 31;
      xs[r * 36 + c] = x[(size_t)(m0 + r) * DIM + kc + c];
    }
    #pragma unroll
    for (int j = 0; j < 16; ++j) {
      int i = t + 256 * j;
      int r = i >> 7, c = i & 127;
      wsd[r * 132 + c] = W_enc[(size_t)(kc + r) * HID + n0 + c];
    }
#endif
    __syncthreads();

    #pragma unroll
    for (int kk = 0; kk < 32; kk += 4) {
      v2f a, b0, b1;
      a.x  = xs[am * 36 + kk + kbase];
      a.y  = xs[am * 36 + kk + kbase + 1];
      b0.x = wsd[(kk + kbase) * 132 + bn0];
      b0.y = wsd[(kk + kbase + 1) * 132 + bn0];
      b1.x = wsd[(kk + kbase) * 132 + bn1];
      b1.y = wsd[(kk + kbase + 1) * 132 + bn1];
      acc0 = __builtin_amdgcn_wmma_f32_16x16x4_f32(false, a, false, b0,
                                                   (short)0, acc0, false, false);
      acc1 = __builtin_amdgcn_wmma_f32_16x16x4_f32(false, a, false, b1,
                                                   (short)0, acc1, false, false);
    }
    __syncthreads();
  }

  // C/D layout: VGPR r -> M=r (lanes 0-15), M=r+8 (lanes 16-31); N = lane%16
  const size_t col0  = (size_t)n0 + bn0;
  const size_t col1  = (size_t)n0 + bn1;
  const float  be0   = b_enc[col0];
  const float  be1   = b_enc[col1];
  const int    rbase = (lane >> 4) * 8;
  #pragma unroll
  for (int r = 0; r < 8; ++r) {
    size_t row = (size_t)m0 + 16 * wm + rbase + r;
    __builtin_nontemporal_store(acc0[r] + be0, &pre[row * HID + col0]);
    __builtin_nontemporal_store(acc1[r] + be1, &pre[row * HID + col1]);
  }
}

// ---------------------------------------------------------------------------
// Kernel 2: inv_norm[h] = 1 / max(||W_dec[h,:]||, 1e-12)   (one wave per row)
// ---------------------------------------------------------------------------
__global__ __launch_bounds__(256)
void sae_dec_norms(const float* __restrict__ W_dec,
                   float* __restrict__ inv_norm) {
  const int lane = threadIdx.x & 31;
  const int wave = threadIdx.x >> 5;
  const int row  = blockIdx.x * 8 + wave;
  const float* wr = W_dec + (size_t)row * DIM;
  float s = 0.f;
  #pragma unroll
  for (int j = 0; j < DIM / 32; ++j) {
    float v = wr[lane + 32 * j];
    s = fmaf(v, v, s);
  }
  #pragma unroll
  for (int off = 16; off > 0; off >>= 1) s += __shfl_xor(s, off, 32);
  if (lane == 0) inv_norm[row] = 1.0f / fmaxf(sqrtf(s), 1e-12f);
}

// ---------------------------------------------------------------------------
// Kernel 3: per-row top-32. Row (64KB) staged into LDS via async-to-LDS
// (320KB/WGP on CDNA5), iterative argmax, dense outputs zero-filled with
// non-temporal stores (1GB streamed, never re-read).
// ---------------------------------------------------------------------------
__global__ __launch_bounds__(256)
void sae_topk(const float* __restrict__ pre,
              float* __restrict__ activated,
              float* __restrict__ mask,
              int* __restrict__ idx_out,
              float* __restrict__ val_out) {
  __shared__ float vals[HID];
  __shared__ float rv[256];
  __shared__ int   ri[256];
  __shared__ int   sel_i[TOPK];
  __shared__ float sel_v[TOPK];

  const int    t    = threadIdx.x;
  const size_t row  = blockIdx.x;
  const float* prow = pre + row * HID;
  float* arow = activated + row * HID;
  float* mrow = mask + row * HID;

#if USE_ASYNC_LDS
  // 16384 floats = 4096 x 16B chunks, 16 async b128 per thread
  #pragma unroll
  for (int j = 0; j < 16; ++j) {
    int c4 = (t + 256 * j) * 4;
    async_copy16(prow + c4, &vals[c4]);
  }
#endif
  // zero-fill dense outputs while the async copies are in flight
  #pragma unroll
  for (int j = 0; j < HID / 256; ++j) {
    int i = t + 256 * j;
#if !USE_ASYNC_LDS
    vals[i] = __builtin_nontemporal_load(&prow[i]);
#endif
    __builtin_nontemporal_store(0.0f, &arow[i]);
    __builtin_nontemporal_store(0.0f, &mrow[i]);
  }
#if USE_ASYNC_LDS
  async_wait0();
#endif
  __syncthreads();

  for (int it = 0; it < TOPK; ++it) {
    float bv = -3.402823466e38f;
    int   bi = 0;
    #pragma unroll
    for (int j = 0; j < HID / 256; ++j) {
      int i = t + 256 * j;
      float v = vals[i];
      if (v > bv) { bv = v; bi = i; }
    }
    rv[t] = bv; ri[t] = bi;
    __syncthreads();
    for (int s = 128; s > 0; s >>= 1) {
      if (t < s && rv[t + s] > rv[t]) { rv[t] = rv[t + s]; ri[t] = ri[t + s]; }
      __syncthreads();
    }
    if (t == 0) {
      sel_i[it] = ri[0];
      sel_v[it] = rv[0];
      vals[ri[0]] = -3.402823466e38f;   // exclude from next rounds
    }
    __syncthreads();
  }

  // scatter the 32 winners (zero-fill completed before the barriers above)
  if (t < TOPK) {
    int   i = sel_i[t];
    float v = sel_v[t];
    idx_out[row * TOPK + t] = i;
    val_out[row * TOPK + t] = v;
    arow[i] = v;
    mrow[i] = (v > 0.0f) ? 1.0f : 0.0f;
  }
}

// ---------------------------------------------------------------------------
// Kernel 4: sparse decode  rec[n,:] = sum_k val*inv_norm[idx] * W_dec[idx,:]
// + b_dec.  32 nonzeros/row -> 512x less work than a dense GEMM.
// W_dec rows are gathered through the cache (hot features get reused);
// rec is a pure output stream -> non-temporal stores.
// ---------------------------------------------------------------------------
__global__ __launch_bounds__(256)
void sae_decode(const int* __restrict__ idx_in,
                const float* __restrict__ val_in,
                const float* __restrict__ inv_norm,
                const float* __restrict__ W_dec,
                const float* __restrict__ b_dec,
                float* __restrict__ rec,
                float* __restrict__ ghost) {
  __shared__ int   si[TOPK];
  __shared__ float sv[TOPK];
  const int    t   = threadIdx.x;
  const size_t row = blockIdx.x;

  if (t < TOPK) {
    int i = idx_in[row * TOPK + t];
    si[t] = i;
    sv[t] = val_in[row * TOPK + t] * inv_norm[i];
  }
  __syncthreads();

  float a0 = 0.f, a1 = 0.f, a2 = 0.f;
  #pragma unroll
  for (int k = 0; k < TOPK; ++k) {
    const float* wr = W_dec + (size_t)si[k] * DIM;
    float s = sv[k];
    a0 = fmaf(s, wr[t], a0);
    a1 = fmaf(s, wr[t + 256], a1);
    a2 = fmaf(s, wr[t + 512], a2);
  }
  __builtin_nontemporal_store(a0 + b_dec[t],       &rec[row * DIM + t]);
  __builtin_nontemporal_store(a1 + b_dec[t + 256], &rec[row * DIM + t + 256]);
  __builtin_nontemporal_store(a2 + b_dec[t + 512], &rec[row * DIM + t + 512]);

  if (row == 0 && t == 0) ghost[0] = 0.0f;  // eval mode: ghost loss == 0
}

// ---------------------------------------------------------------------------
extern "C" void kernel_launch(void* const* d_in, const int* in_sizes, int n_in,
                              void* d_out, int out_size, void* d_ws, size_t ws_size,
                              hipStream_t stream) {
  const float* x     = (const float*)d_in[0];
  const float* W_enc = (const float*)d_in[1];
  const float* b_enc = (const float*)d_in[2];
  const float* W_dec = (const float*)d_in[3];
  const float* b_dec = (const float*)d_in[4];

  float* out = (float*)d_out;
  const size_t ACT = (size_t)NROWS * DIM;                  // after reconstruction
  const size_t PRE = ACT + (size_t)NROWS * HID;            // after activated
  const size_t MSK = PRE + (size_t)NROWS * HID;            // after pre_activation
  const size_t GST = MSK + (size_t)NROWS * HID;            // ghost_loss scalar

  float* rec   = out;
  float* act   = out + ACT;
  float* pre   = out + PRE;
  float* mask  = out + MSK;
  float* ghost = out + GST;

  // workspace: inv_norm[16384] | idx[8192*32] | val[8192*32]
  float* inv_norm = (float*)d_ws;
  int*   idx_ws   = (int*)((char*)d_ws + 64 * 1024);
  float* val_ws   = (float*)((char*)d_ws + 64 * 1024 + (size_t)NROWS * TOPK * 4);

  dim3 gEnc(HID / 128, NROWS / 32);   // 128 x 256 blocks
  sae_encode_wmma<<<gEnc, 256, 0, stream>>>(x, W_enc, b_enc, pre);
  sae_dec_norms<<<HID / 8, 256, 0, stream>>>(W_dec, inv_norm);
  sae_topk<<<NROWS, 256, 0, stream>>>(pre, act, mask, idx_ws, val_ws);
  sae_decode<<<NROWS, 256, 0, stream>>>(idx_ws, val_ws, inv_norm, W_dec, b_dec,
                                        rec, ghost);
}